// FP4Quantizer_35459249996635
// MI455X (gfx1250) — compile-verified
//
#include <hip/hip_runtime.h>
#include <stdint.h>

// FP4 quant-dequant for 4096x4096 fp32 (memory-bound; ~128MB HBM traffic).
// Pass 1: per-64-elem-block 95th percentile of |x| via per-thread top-5
//         selection. The 32KB input tile is staged into LDS by ONE Tensor
//         Data Mover op per workgroup (tensor_load_to_lds), whose descriptor
//         pad fields re-tile into padded 272B rows and whose 1-D tensor_dim
//         gives reference-faithful zero-fill at the tail. Completion via
//         s_wait_tensorcnt. Global scale min/max via wave32 shuffle reduce +
//         uint-keyed float atomics.
// Pass 2: streaming float4 quantize->dequantize (x stays resident in the
//         192MB L2 between passes).

#define QBLOCK 64            // quantization block size (elements)
#define BLKS_PER_WG 128      // one quant block per thread
#define ROW_F 68             // 64 floats + 4-float pad (TDM pad_amount=3 -> 4 DWORDs)
#define TILE_ELEMS (BLKS_PER_WG * QBLOCK)   // 8192 floats per WG

typedef uint32_t u32x4 __attribute__((ext_vector_type(4)));
typedef uint32_t u32x8 __attribute__((ext_vector_type(8)));

// insert |v| into descending top-5 (m0>=...>=m4) with a min/max bubble chain
#define INS5(vv)                                              \
  do {                                                        \
    float c_ = fabsf(vv), t_;                                 \
    t_ = fminf(m0, c_); m0 = fmaxf(m0, c_); c_ = t_;          \
    t_ = fminf(m1, c_); m1 = fmaxf(m1, c_); c_ = t_;          \
    t_ = fminf(m2, c_); m2 = fmaxf(m2, c_); c_ = t_;          \
    t_ = fminf(m3, c_); m3 = fmaxf(m3, c_); c_ = t_;          \
    m4 = fmaxf(m4, c_);                                       \
  } while (0)

__global__ __launch_bounds__(BLKS_PER_WG) void fp4_scales_kernel(
    const float* __restrict__ x, float* __restrict__ scales,
    unsigned int* __restrict__ minmax, int nblocks, long long n) {
  __shared__ float tile[BLKS_PER_WG * ROW_F];

  const int t = threadIdx.x;
  const long long wgFloatBase = (long long)blockIdx.x * TILE_ELEMS;

  // ---- TDM: one DMA per workgroup, issued by wave 0 only. readfirstlane
  // forces a scalar (uniform) branch: TDM ignores EXEC, so waves 1-3 must
  // branch around it, not execute it with EXEC=0.
  const uint32_t waveId = __builtin_amdgcn_readfirstlane((uint32_t)t >> 5);
  if (waveId == 0) {
    const uint64_t gaddr = (uint64_t)(uintptr_t)(x + wgFloatBase);
    const uint32_t lds0  = (uint32_t)(uintptr_t)(&tile[0]); // low 32b = LDS offset
    const uint32_t td0   = (uint32_t)(n - wgFloatBase);     // 1-D tensor -> OOB reads 0

    // D# group 0: count=1 | lds_addr | global_addr[56:0] | type=2 (bits 127:126)
    const u32x4 g0 = {
        1u,
        lds0,
        (uint32_t)gaddr,
        (uint32_t)((gaddr >> 32) & 0x01FFFFFFull) | 0x80000000u};

    // D# group 1:
    //  w0: data_size=2(4B)<<16 | pad_enable<<20 | pad_interval=5<<22 (64 DW)
    //      | pad_amount=3<<25 (4 DW)  => padded 272B LDS rows
    //  tensor_dim0 = td0 (bits 79:48), tensor_dim1 = 1 (bits 111:80)
    //  tile_dim0 = 8192 (bits 127:112), tile_dim1 = 1 (bits 143:128)
    //  tensor_dim0_stride = 8192 (bits 207:160, unused with one row)
    const u32x8 g1 = {
        0x07520000u,
        (td0 & 0xFFFFu) << 16,
        ((td0 >> 16) & 0xFFFFu) | (1u << 16),
        (uint32_t)TILE_ELEMS << 16,
        1u,
        (uint32_t)TILE_ELEMS,
        0u,
        0u};

    asm volatile("tensor_load_to_lds %0, %1" :: "s"(g0), "s"(g1) : "memory");
    __builtin_amdgcn_s_wait_tensorcnt(0);
  }
  __syncthreads();   // publish LDS tile to all waves

  // top-5 |.| over this thread's 64-value block (abs values >= 0, so -1 sentinel)
  float m0 = -1.f, m1 = -1.f, m2 = -1.f, m3 = -1.f, m4 = -1.f;
  const float4* row = (const float4*)&tile[t * ROW_F];
#pragma unroll
  for (int j = 0; j < QBLOCK / 4; ++j) {
    float4 v = row[j];
    INS5(v.x); INS5(v.y); INS5(v.z); INS5(v.w);
  }
  // q = 0.95*(64-1) = 59.85 -> s = v[59] + 0.85*(v[60]-v[59]); m3=4th, m4=5th largest
  float scale = fmaxf(m4 + 0.85f * (m3 - m4), 1e-8f);

  const int gblock = blockIdx.x * BLKS_PER_WG + t;
  const bool valid = gblock < nblocks;
  if (valid) scales[gblock] = scale;

  // wave32 min/max reduce, then one pair of atomics per wave (uint trick: s > 0)
  float lo = valid ? scale : __uint_as_float(0x7F800000u);
  float hi = valid ? scale : 0.0f;
#pragma unroll
  for (int m = 16; m >= 1; m >>= 1) {
    lo = fminf(lo, __shfl_xor(lo, m, 32));
    hi = fmaxf(hi, __shfl_xor(hi, m, 32));
  }
  if ((t & 31) == 0) {
    atomicMin(&minmax[0], __float_as_uint(lo));
    atomicMax(&minmax[1], __float_as_uint(hi));
  }
}

__global__ void init_minmax_kernel(unsigned int* minmax) {
  minmax[0] = 0x7F800000u;  // +inf bits (running min of positive floats)
  minmax[1] = 0u;           // +0.0 bits (running max)
}

// nearest level of [-3,-2,-1.5,-1,-0.75,0,0.75,1,1.5,2,3] via symmetric midpoints
__device__ __forceinline__ float fp4_level(float xn) {
  float a = fabsf(xn);
  float q = a < 0.375f ? 0.0f
          : a < 0.875f ? 0.75f
          : a < 1.25f  ? 1.0f
          : a < 1.75f  ? 1.5f
          : a < 2.5f   ? 2.0f
          :              3.0f;
  return copysignf(q, xn);
}

__global__ __launch_bounds__(256) void fp4_dequant_kernel(
    const float4* __restrict__ x4, const float* __restrict__ scales,
    const unsigned int* __restrict__ minmax, float4* __restrict__ out4, int n4) {
  const int i = blockIdx.x * blockDim.x + threadIdx.x;
  if (i >= n4) return;

  const float smin = __uint_as_float(minmax[0]);   // uniform -> scalar load
  const float smax = __uint_as_float(minmax[1]);
  const float s = scales[i >> 4];  // 16 float4-threads per quant block

  float dq;
  if (smax > smin) {
    float ss = (smax - smin) * (1.0f / 255.0f);
    float q  = fminf(fmaxf(rintf((s - smin) / ss), 0.0f), 255.0f);
    dq = q * ss;                   // smin intentionally dropped (matches source)
  } else {
    dq = 0.0f;                     // q=0, ss=1 -> deq scale 0
  }

  const float inv_s = 1.0f / s;
  float4 v = x4[i];
  float4 o;
  o.x = fp4_level(v.x * inv_s) * dq;
  o.y = fp4_level(v.y * inv_s) * dq;
  o.z = fp4_level(v.z * inv_s) * dq;
  o.w = fp4_level(v.w * inv_s) * dq;
  out4[i] = o;
}

extern "C" void kernel_launch(void* const* d_in, const int* in_sizes, int n_in,
                              void* d_out, int out_size, void* d_ws, size_t ws_size,
                              hipStream_t stream) {
  const float* x = (const float*)d_in[0];
  const long long n = (long long)in_sizes[0];           // 4096*4096
  const int nblocks = (int)((n + QBLOCK - 1) / QBLOCK); // 262,144

  unsigned int* minmax = (unsigned int*)d_ws;                 // 2 uints
  float* scales = (float*)((char*)d_ws + 16);                 // 1MB of scales

  hipLaunchKernelGGL(init_minmax_kernel, dim3(1), dim3(1), 0, stream, minmax);

  const int wgs = (nblocks + BLKS_PER_WG - 1) / BLKS_PER_WG;  // 2048
  hipLaunchKernelGGL(fp4_scales_kernel, dim3(wgs), dim3(BLKS_PER_WG), 0, stream,
                     x, scales, minmax, nblocks, n);

  const int n4 = (int)(n / 4);
  hipLaunchKernelGGL(fp4_dequant_kernel, dim3((n4 + 255) / 256), dim3(256), 0,
                     stream, (const float4*)x, scales, minmax, (float4*)d_out, n4);
}